// TransformerBlock_68719477153
// MI455X (gfx1250) — compile-verified
//
#include <hip/hip_runtime.h>

#define BATCH 131072
#define TT 7
#define DD 3
#define ROWS (BATCH * TT)      // 917504
#define TILES (ROWS / 16)      // 57344
#define SCALEQK 0.35355339059327373f
#define EPS 1e-6f

typedef float v2f __attribute__((ext_vector_type(2)));
typedef float v8f __attribute__((ext_vector_type(8)));

#if __has_builtin(__builtin_amdgcn_tanhf)
#define FAST_TANH(x) __builtin_amdgcn_tanhf(x)
#else
#define FAST_TANH(x) tanhf(x)
#endif

__device__ __forceinline__ float fast_rcp(float x) { return __builtin_amdgcn_rcpf(x); }
__device__ __forceinline__ float fast_rsq(float x) { return __builtin_amdgcn_rsqf(x); }

__device__ __forceinline__ void ln3(float s0, float s1, float s2,
                                    float g0, float g1, float g2,
                                    float b0, float b1, float b2,
                                    float* o) {
    float mu = (s0 + s1 + s2) * (1.0f / 3.0f);
    float d0 = s0 - mu, d1 = s1 - mu, d2 = s2 - mu;
    float var = (d0 * d0 + d1 * d1 + d2 * d2) * (1.0f / 3.0f);
    float inv = fast_rsq(var + EPS);
    o[0] = d0 * inv * g0 + b0;
    o[1] = d1 * inv * g1 + b1;
    o[2] = d2 * inv * g2 + b2;
}

// ---------------------------------------------------------------------------
// Kernel 1: attention (collapsed) + LN1 + RNN x2 (+LN1 each) -> ws[ROWS][3]
// One thread per batch element.
// ---------------------------------------------------------------------------
__global__ __launch_bounds__(256) void attn_rnn_kernel(
    const float* __restrict__ x,
    const float* __restrict__ wq, const float* __restrict__ bq,
    const float* __restrict__ wk, const float* __restrict__ bk,
    const float* __restrict__ wv, const float* __restrict__ bv,
    const float* __restrict__ wo, const float* __restrict__ bo,
    const float* __restrict__ ln1g, const float* __restrict__ ln1b,
    const float* __restrict__ rnnk, const float* __restrict__ rnnr,
    const float* __restrict__ rnnb,
    float* __restrict__ ws) {
    // Derived, weight-only attention matrices (per head):
    //   A[d][e] = sum_k wq[d][h][k] wk[e][h][k]      (q.k core)
    //   M[d][e] = sum_k wv[d][h][k] wo[h][k][e]      (v->out core)
    //   G[d] = wq[d][h][:].bk ; F[e] = bq.wk[e][h][:] ; C0 = bq.bk
    //   M0[d] = bv.wo[h][:][d]
    __shared__ float sA[3][9], sM[3][9], sG[3][3], sF[3][3], sM0[3][3], sC0[3];

    if (threadIdx.x == 0) {
        for (int h = 0; h < 3; ++h) {
            for (int d = 0; d < 3; ++d)
                for (int e = 0; e < 3; ++e) {
                    float a = 0.f, m = 0.f;
                    for (int k = 0; k < 8; ++k) {
                        a += wq[(d * 3 + h) * 8 + k] * wk[(e * 3 + h) * 8 + k];
                        m += wv[(d * 3 + h) * 8 + k] * wo[(h * 8 + k) * 3 + e];
                    }
                    sA[h][d * 3 + e] = a;
                    sM[h][d * 3 + e] = m;
                }
            for (int d = 0; d < 3; ++d) {
                float gg = 0.f, ff = 0.f, mm = 0.f;
                for (int k = 0; k < 8; ++k) {
                    gg += wq[(d * 3 + h) * 8 + k] * bk[h * 8 + k];
                    ff += bq[h * 8 + k] * wk[(d * 3 + h) * 8 + k];
                    mm += bv[h * 8 + k] * wo[(h * 8 + k) * 3 + d];
                }
                sG[h][d] = gg; sF[h][d] = ff; sM0[h][d] = mm;
            }
            float c = 0.f;
            for (int k = 0; k < 8; ++k) c += bq[h * 8 + k] * bk[h * 8 + k];
            sC0[h] = c;
        }
    }
    __syncthreads();

    int b = blockIdx.x * blockDim.x + threadIdx.x;
    if (b >= BATCH) return;

    float xs[7][3];
    const float* xb = x + (size_t)b * 21;
#pragma unroll
    for (int t = 0; t < 7; ++t)
#pragma unroll
        for (int d = 0; d < 3; ++d) xs[t][d] = xb[t * 3 + d];

    float bo0 = bo[0], bo1 = bo[1], bo2 = bo[2];
    float ao[7][3];
#pragma unroll
    for (int t = 0; t < 7; ++t) { ao[t][0] = bo0; ao[t][1] = bo1; ao[t][2] = bo2; }

    for (int h = 0; h < 3; ++h) {
        float u[7][3], P[7][3], av[7], bv2[7];
#pragma unroll
        for (int t = 0; t < 7; ++t) {
#pragma unroll
            for (int j = 0; j < 3; ++j) {
                u[t][j] = xs[t][0] * sA[h][0 * 3 + j] + xs[t][1] * sA[h][1 * 3 + j] +
                          xs[t][2] * sA[h][2 * 3 + j];
                P[t][j] = xs[t][0] * sM[h][0 * 3 + j] + xs[t][1] * sM[h][1 * 3 + j] +
                          xs[t][2] * sM[h][2 * 3 + j] + sM0[h][j];
            }
            av[t] = xs[t][0] * sG[h][0] + xs[t][1] * sG[h][1] + xs[t][2] * sG[h][2];
            bv2[t] = xs[t][0] * sF[h][0] + xs[t][1] * sF[h][1] + xs[t][2] * sF[h][2];
        }
        float c0 = sC0[h];
#pragma unroll
        for (int t = 0; t < 7; ++t) {
            float sc[7], mx = -3.0e38f;
#pragma unroll
            for (int s = 0; s < 7; ++s) {
                float v = (u[t][0] * xs[s][0] + u[t][1] * xs[s][1] + u[t][2] * xs[s][2] +
                           av[t] + bv2[s] + c0) * SCALEQK;
                sc[s] = v;
                mx = fmaxf(mx, v);
            }
            float sum = 0.f;
#pragma unroll
            for (int s = 0; s < 7; ++s) { sc[s] = __expf(sc[s] - mx); sum += sc[s]; }
            float inv = fast_rcp(sum);
#pragma unroll
            for (int d = 0; d < 3; ++d) {
                float acc = 0.f;
#pragma unroll
                for (int s = 0; s < 7; ++s) acc += sc[s] * P[s][d];
                ao[t][d] += acc * inv;
            }
        }
    }

    float lg0 = ln1g[0], lg1 = ln1g[1], lg2 = ln1g[2];
    float lb0 = ln1b[0], lb1 = ln1b[1], lb2 = ln1b[2];

    float o1[7][3];
#pragma unroll
    for (int t = 0; t < 7; ++t)
        ln3(xs[t][0] + ao[t][0], xs[t][1] + ao[t][1], xs[t][2] + ao[t][2],
            lg0, lg1, lg2, lb0, lb1, lb2, o1[t]);

    float rk[9], rr[9], rb3[3];
#pragma unroll
    for (int i = 0; i < 9; ++i) { rk[i] = rnnk[i]; rr[i] = rnnr[i]; }
#pragma unroll
    for (int i = 0; i < 3; ++i) rb3[i] = rnnb[i];

    auto rnn_pass = [&](float in[7][3], float outp[7][3]) {
        float h0 = 0.f, h1 = 0.f, h2 = 0.f;
#pragma unroll
        for (int t = 0; t < 7; ++t) {
            float p0 = in[t][0] * rk[0] + in[t][1] * rk[3] + in[t][2] * rk[6] + rb3[0];
            float p1 = in[t][0] * rk[1] + in[t][1] * rk[4] + in[t][2] * rk[7] + rb3[1];
            float p2 = in[t][0] * rk[2] + in[t][1] * rk[5] + in[t][2] * rk[8] + rb3[2];
            float n0 = FAST_TANH(p0 + h0 * rr[0] + h1 * rr[3] + h2 * rr[6]);
            float n1 = FAST_TANH(p1 + h0 * rr[1] + h1 * rr[4] + h2 * rr[7]);
            float n2 = FAST_TANH(p2 + h0 * rr[2] + h1 * rr[5] + h2 * rr[8]);
            h0 = n0; h1 = n1; h2 = n2;
            outp[t][0] = n0; outp[t][1] = n1; outp[t][2] = n2;
        }
    };

    float r1[7][3], o2[7][3], r2[7][3], o21[7][3];
    rnn_pass(o1, r1);
#pragma unroll
    for (int t = 0; t < 7; ++t)
        ln3(o1[t][0] + r1[t][0], o1[t][1] + r1[t][1], o1[t][2] + r1[t][2],
            lg0, lg1, lg2, lb0, lb1, lb2, o2[t]);
    rnn_pass(o2, r2);
#pragma unroll
    for (int t = 0; t < 7; ++t)
        ln3(o2[t][0] + r2[t][0], o2[t][1] + r2[t][1], o2[t][2] + r2[t][2],
            lg0, lg1, lg2, lb0, lb1, lb2, o21[t]);

    float* wp = ws + (size_t)b * 21;
#pragma unroll
    for (int t = 0; t < 7; ++t)
#pragma unroll
        for (int d = 0; d < 3; ++d) wp[t * 3 + d] = o21[t][d];
}

// ---------------------------------------------------------------------------
// Kernel 2: FFN via v_wmma_f32_16x16x4_f32 + residual + LN2 -> out[ROWS][3]
// One 16-row tile per wave, grid-stride over tiles.
// ---------------------------------------------------------------------------
#define HSTR 68   // padded H row stride (floats) -> conflict-free ds loads

__global__ __launch_bounds__(256) void ffn_wmma_kernel(
    const float* __restrict__ xin,   // ws [ROWS][3]
    const float* __restrict__ d1w,   // [3][64]
    const float* __restrict__ d1b,   // [64]
    const float* __restrict__ d2w,   // [64][3]
    const float* __restrict__ d2b,   // [3]
    const float* __restrict__ lng,   // [3]
    const float* __restrict__ lnb,   // [3]
    float* __restrict__ out) {
    __shared__ float lds_h[8 * 16 * HSTR];  // per-wave 16x64 (padded) H tile
    __shared__ float lds_y[8 * 16 * 4];     // per-wave 16x3 (padded) Y tile

    const int lane = threadIdx.x & 31;
    const int wave = threadIdx.x >> 5;
    const int col = lane & 15;
    const bool hi = lane >= 16;
    float* Hw = lds_h + wave * 16 * HSTR;
    float* Yw = lds_y + wave * 16 * 4;

    // B1 (d1_w, K=4 pad, N in 4 tiles of 16): v0 = row K0(lo)/K2(hi), v1 = K1(lo)/K3=0(hi)
    v2f b1[4];
    float hb[4];
#pragma unroll
    for (int n = 0; n < 4; ++n) {
        int c = n * 16 + col;
        float v0 = d1w[(hi ? 2 : 0) * 64 + c];
        float v1 = hi ? 0.0f : d1w[64 + c];
        b1[n] = (v2f){v0, v1};
        hb[n] = d1b[c];
    }
    // B2 (d2_w, 16 K-steps of 4, N padded 3->16)
    v2f b2[16];
#pragma unroll
    for (int kk = 0; kk < 16; ++kk) {
        int k0 = 4 * kk + (hi ? 2 : 0);
        float v0 = (col < 3) ? d2w[k0 * 3 + col] : 0.0f;
        float v1 = (col < 3) ? d2w[(k0 + 1) * 3 + col] : 0.0f;
        b2[kk] = (v2f){v0, v1};
    }

    float g0 = lng[0], g1 = lng[1], g2 = lng[2];
    float bb0 = lnb[0], bb1 = lnb[1], bb2 = lnb[2];
    float db0 = d2b[0], db1 = d2b[1], db2 = d2b[2];

    const int nwaves = gridDim.x << 3;
    for (int tile = blockIdx.x * 8 + wave; tile < TILES; tile += nwaves) {
        const int row0 = tile * 16;

        // A1: lane<16 -> X[col][0..1], lane>=16 -> X[col][2], 0 (K=3 pad)
        const float* xr = xin + (size_t)(row0 + col) * 3;
        v2f aA;
        if (!hi) aA = (v2f){xr[0], xr[1]};
        else     aA = (v2f){xr[2], 0.0f};

        // GEMM1: H[16][64] = X16 * d1_w  (4 WMMAs over N tiles)
        v8f hacc[4];
#pragma unroll
        for (int n = 0; n < 4; ++n) {
            v8f cz = {};
            hacc[n] = __builtin_amdgcn_wmma_f32_16x16x4_f32(
                false, aA, false, b1[n], (short)0, cz, false, false);
        }
        // bias + relu, store to LDS in row-major (padded) for A re-layout
#pragma unroll
        for (int n = 0; n < 4; ++n) {
            int c = n * 16 + col;
#pragma unroll
            for (int j = 0; j < 8; ++j) {
                int r = hi ? (8 + j) : j;
                Hw[r * HSTR + c] = fmaxf(hacc[n][j] + hb[n], 0.0f);
            }
        }

        // GEMM2: Y[16][16] = H * d2_w  (K=64 -> 16 chained WMMAs)
        v8f acc = {};
        const float* hpbase = Hw + col * HSTR + (hi ? 2 : 0);
#pragma unroll
        for (int kk = 0; kk < 16; ++kk) {
            const v2f a2 = *(const v2f*)(hpbase + 4 * kk);
            acc = __builtin_amdgcn_wmma_f32_16x16x4_f32(
                false, a2, false, b2[kk], (short)0, acc, false, false);
        }

        // Scatter valid Y columns (0..2) to LDS
        if (col < 3) {
            int rb = hi ? 8 : 0;
#pragma unroll
            for (int j = 0; j < 8; ++j) Yw[(rb + j) * 4 + col] = acc[j];
        }

        // Finalize one row per lane (lanes 0..15): +d2_b, residual, LN2
        if (!hi) {
            int r = row0 + lane;
            const float* xp = xin + (size_t)r * 3;
            float s0 = xp[0] + Yw[lane * 4 + 0] + db0;
            float s1 = xp[1] + Yw[lane * 4 + 1] + db1;
            float s2 = xp[2] + Yw[lane * 4 + 2] + db2;
            float o[3];
            ln3(s0, s1, s2, g0, g1, g2, bb0, bb1, bb2, o);
            float* op = out + (size_t)r * 3;
            op[0] = o[0]; op[1] = o[1]; op[2] = o[2];
        }
    }
}

extern "C" void kernel_launch(void* const* d_in, const int* in_sizes, int n_in,
                              void* d_out, int out_size, void* d_ws, size_t ws_size,
                              hipStream_t stream) {
    const float* x     = (const float*)d_in[0];
    const float* wq    = (const float*)d_in[1];
    const float* bq    = (const float*)d_in[2];
    const float* wk    = (const float*)d_in[3];
    const float* bk    = (const float*)d_in[4];
    const float* wv    = (const float*)d_in[5];
    const float* bv    = (const float*)d_in[6];
    const float* wo    = (const float*)d_in[7];
    const float* bo    = (const float*)d_in[8];
    const float* ln1g  = (const float*)d_in[9];
    const float* ln1b  = (const float*)d_in[10];
    const float* rnnk  = (const float*)d_in[11];
    const float* rnnr  = (const float*)d_in[12];
    const float* rnnb  = (const float*)d_in[13];
    const float* d1w   = (const float*)d_in[14];
    const float* d1b   = (const float*)d_in[15];
    const float* d2w   = (const float*)d_in[16];
    const float* d2b   = (const float*)d_in[17];
    const float* ln2g  = (const float*)d_in[18];
    const float* ln2b  = (const float*)d_in[19];

    float* out = (float*)d_out;
    float* ws  = (float*)d_ws;   // ROWS*3 floats = ~11 MB

    attn_rnn_kernel<<<BATCH / 256, 256, 0, stream>>>(
        x, wq, bq, wk, bk, wv, bv, wo, bo, ln1g, ln1b, rnnk, rnnr, rnnb, ws);
    ffn_wmma_kernel<<<2048, 256, 0, stream>>>(
        ws, d1w, d1b, d2w, d2b, ln2g, ln2b, out);
}